// HypergraphModel_56642028700408
// MI455X (gfx1250) — compile-verified
//
#include <hip/hip_runtime.h>
#include <hip/hip_bf16.h>

// ---------------------------------------------------------------------------
// Hypergraph model pipeline for MI455X (gfx1250, wave32).
// Dense GEMMs use V_WMMA_F32_16X16X4_F32. Each wave computes a 16 x (16*NT)
// strip so the A fragment (the bandwidth-dominant operand) is loaded ONCE and
// fed to NT WMMAs. Scatters use global_atomic_add_f32.
// ---------------------------------------------------------------------------

typedef float v2f __attribute__((ext_vector_type(2)));
typedef float v8f __attribute__((ext_vector_type(8)));

// ---------------------------------------------------------------------------
// Generic f32 WMMA GEMM: D[m, n] = op( sum_k opA(A[m,k]) * B[k,n] + bias[n] )
// One wave computes a 16 x (16*NT) strip (NT accumulators, shared A fragment).
// Grid: (ceil(M/16), Ntiles/NT), block: 32.  Requires K % 4 == 0.
// Fragment layouts per CDNA5 ISA:
//   A 16x4 f32: lane r in [0,15] -> row r, vgpr {K+0,K+1}; lanes 16..31 -> {K+2,K+3}
//   B 4x16 f32: lane r -> col r, same K split by half-wave
//   C/D 16x16 f32: lane -> col, vgpr i -> row i (+8 for upper half-wave)
// ---------------------------------------------------------------------------
template <int NT, bool RELU_A, bool HAS_BIAS, bool RELU_D>
__global__ void wmma_gemm_f32_kernel(const float* __restrict__ A, int lda,
                                     const float* __restrict__ B, int ldb,
                                     const float* __restrict__ bias,
                                     float* __restrict__ D, int ldd,
                                     int M, int K) {
  const int tm   = blockIdx.x;
  const int tn0  = blockIdx.y * NT;
  const int lane = threadIdx.x;
  const int r    = lane & 15;
  const int hi   = lane >> 4;       // half-wave select
  const int kh   = hi << 1;         // K sub-offset: 0 or 2

  int arow = tm * 16 + r;
  if (arow >= M) arow = M - 1;      // clamp loads; stores are guarded
  const float* __restrict__ Arow = A + (long)arow * lda;

  const float* Bcol[NT];
#pragma unroll
  for (int nt = 0; nt < NT; ++nt) Bcol[nt] = B + (tn0 + nt) * 16 + r;

  v8f acc[NT];
#pragma unroll
  for (int nt = 0; nt < NT; ++nt) acc[nt] = (v8f){};

  for (int k = 0; k < K; k += 4) {
    v2f a;
    a.x = Arow[k + kh];
    a.y = Arow[k + kh + 1];
    if (RELU_A) { a.x = fmaxf(a.x, 0.0f); a.y = fmaxf(a.y, 0.0f); }
#pragma unroll
    for (int nt = 0; nt < NT; ++nt) {
      v2f b;
      b.x = Bcol[nt][(long)(k + kh) * ldb];
      b.y = Bcol[nt][(long)(k + kh + 1) * ldb];
      acc[nt] = __builtin_amdgcn_wmma_f32_16x16x4_f32(false, a, false, b,
                                                      (short)0, acc[nt],
                                                      false, false);
    }
  }

#pragma unroll
  for (int nt = 0; nt < NT; ++nt) {
    const float bv = HAS_BIAS ? bias[(tn0 + nt) * 16 + r] : 0.0f;
#pragma unroll
    for (int i = 0; i < 8; ++i) {
      const int orow = tm * 16 + i + 8 * hi;
      if (orow < M) {
        float v = acc[nt][i] + bv;
        if (RELU_D) v = fmaxf(v, 0.0f);
        D[(long)orow * ldd + (tn0 + nt) * 16 + r] = v;
      }
    }
  }
}

// ---------------------------------------------------------------------------
// Fused: SH = relu( T[M,1024] @ EH[1024,32] + (NF*invE)[M,32] @ Wn[32,32] + bg )
// One wave computes the full 16 x 32 output strip (2 accumulators, shared A),
// so T is streamed exactly once.  Branchless clamped prefetch of T ~1KB ahead.
// Grid: (M/16, 1), block: 32.
// ---------------------------------------------------------------------------
__global__ void shared_fused_wmma_kernel(const float* __restrict__ T,
                                         const float* __restrict__ EH,
                                         const float* __restrict__ NF,
                                         const float* __restrict__ Wn,
                                         const float* __restrict__ bg,
                                         float invE,
                                         float* __restrict__ SH, int M) {
  const int tm   = blockIdx.x;
  const int lane = threadIdx.x;
  const int r    = lane & 15;
  const int hi   = lane >> 4;
  const int kh   = hi << 1;

  int arow = tm * 16 + r;
  if (arow >= M) arow = M - 1;
  const float* __restrict__ Trow = T + (long)arow * 1024;
  const float* __restrict__ B0   = EH + r;
  const float* __restrict__ B1   = EH + 16 + r;

  v8f acc0 = {}, acc1 = {};
  for (int k0 = 0; k0 < 1024; k0 += 32) {
    // clamped (always in-bounds) prefetch ~256 floats (1KB) ahead, near-cache
    int pf = k0 + 256; if (pf > 1024 - 32) pf = 1024 - 32;
    __builtin_prefetch(Trow + pf, 0, 3);
#pragma unroll
    for (int k = k0; k < k0 + 32; k += 4) {
      v2f a;
      a.x = Trow[k + kh];
      a.y = Trow[k + kh + 1];
      v2f b0, b1;
      b0.x = B0[(k + kh) * 32];
      b0.y = B0[(k + kh + 1) * 32];
      b1.x = B1[(k + kh) * 32];
      b1.y = B1[(k + kh + 1) * 32];
      acc0 = __builtin_amdgcn_wmma_f32_16x16x4_f32(false, a, false, b0,
                                                   (short)0, acc0, false, false);
      acc1 = __builtin_amdgcn_wmma_f32_16x16x4_f32(false, a, false, b1,
                                                   (short)0, acc1, false, false);
    }
  }

  // + (NF * invE) @ Wn into the same accumulators
  const float* __restrict__ Nrow = NF + (long)arow * 32;
#pragma unroll
  for (int k = 0; k < 32; k += 4) {
    v2f a;
    a.x = Nrow[k + kh] * invE;
    a.y = Nrow[k + kh + 1] * invE;
    v2f b0, b1;
    b0.x = Wn[(k + kh) * 32 + r];
    b0.y = Wn[(k + kh + 1) * 32 + r];
    b1.x = Wn[(k + kh) * 32 + 16 + r];
    b1.y = Wn[(k + kh + 1) * 32 + 16 + r];
    acc0 = __builtin_amdgcn_wmma_f32_16x16x4_f32(false, a, false, b0,
                                                 (short)0, acc0, false, false);
    acc1 = __builtin_amdgcn_wmma_f32_16x16x4_f32(false, a, false, b1,
                                                 (short)0, acc1, false, false);
  }

  const float bv0 = bg[r];
  const float bv1 = bg[16 + r];
#pragma unroll
  for (int i = 0; i < 8; ++i) {
    const int orow = tm * 16 + i + 8 * hi;
    if (orow < M) {
      SH[(long)orow * 32 + r]      = fmaxf(acc0[i] + bv0, 0.0f);
      SH[(long)orow * 32 + 16 + r] = fmaxf(acc1[i] + bv1, 0.0f);
    }
  }
}

// ---------------------------------------------------------------------------
// Edge scatter: out[dst[e], :] += (w ? w[e] : 1) * H[src[e], :]
// F4 = feature_count / 4. One thread handles 4 floats (128-bit gather load,
// 4x global_atomic_add_f32).
// ---------------------------------------------------------------------------
__global__ void edge_scatter_kernel(const float* __restrict__ H,
                                    const int* __restrict__ src,
                                    const int* __restrict__ dst,
                                    const float* __restrict__ w,
                                    float* __restrict__ out,
                                    int F4, int E) {
  const long gid   = (long)blockIdx.x * blockDim.x + threadIdx.x;
  const long total = (long)E * F4;
  if (gid >= total) return;
  const int e  = (int)(gid / F4);
  const int c  = (int)(gid % F4) * 4;
  const int F  = F4 * 4;
  const int s  = src[e];
  const int d  = dst[e];
  const float4 v = *reinterpret_cast<const float4*>(H + (long)s * F + c);
  const float ww = w ? w[e] : 1.0f;
  float* op = out + (long)d * F + c;
  atomicAdd(op + 0, v.x * ww);
  atomicAdd(op + 1, v.y * ww);
  atomicAdd(op + 2, v.z * ww);
  atomicAdd(op + 3, v.w * ww);
}

__global__ void zero_kernel(float* __restrict__ p, long n) {
  long i = (long)blockIdx.x * blockDim.x + threadIdx.x;
  const long stride = (long)gridDim.x * blockDim.x;
  for (; i < n; i += stride) p[i] = 0.0f;
}

__global__ void relu_inplace_kernel(float* __restrict__ p, long n) {
  long i = (long)blockIdx.x * blockDim.x + threadIdx.x;
  const long stride = (long)gridDim.x * blockDim.x;
  for (; i < n; i += stride) p[i] = fmaxf(p[i], 0.0f);
}

// ---------------------------------------------------------------------------
// Per-node classifier: c = relu(shared@Wc1+bc1); logits = c@Wc2+bc2; softmax.
// Weights cached in LDS. One thread per node.
// ---------------------------------------------------------------------------
__global__ void classifier_kernel(const float* __restrict__ SH,
                                  const float* __restrict__ Wc1,
                                  const float* __restrict__ bc1,
                                  const float* __restrict__ Wc2,
                                  const float* __restrict__ bc2,
                                  float* __restrict__ out, int N) {
  __shared__ float sW1[32 * 32];
  __shared__ float sb1[32];
  __shared__ float sW2[32 * 2];
  __shared__ float sb2[2];
  const int t = threadIdx.x;
  for (int i = t; i < 32 * 32; i += blockDim.x) sW1[i] = Wc1[i];
  if (t < 32) sb1[t] = bc1[t];
  if (t < 64) sW2[t] = Wc2[t];
  if (t < 2)  sb2[t] = bc2[t];
  __syncthreads();

  const int n = blockIdx.x * blockDim.x + t;
  if (n >= N) return;

  float s[32];
  const float* __restrict__ row = SH + (long)n * 32;
#pragma unroll
  for (int i = 0; i < 32; ++i) s[i] = row[i];

  float l0 = sb2[0], l1 = sb2[1];
#pragma unroll 4
  for (int j = 0; j < 32; ++j) {
    float c = sb1[j];
#pragma unroll
    for (int i = 0; i < 32; ++i) c = fmaf(s[i], sW1[i * 32 + j], c);
    c = fmaxf(c, 0.0f);
    l0 = fmaf(c, sW2[j * 2 + 0], l0);
    l1 = fmaf(c, sW2[j * 2 + 1], l1);
  }
  const float m  = fmaxf(l0, l1);
  const float e0 = __expf(l0 - m);
  const float e1 = __expf(l1 - m);
  const float inv = 1.0f / (e0 + e1);
  out[(long)n * 2 + 0] = e0 * inv;
  out[(long)n * 2 + 1] = e1 * inv;
}

// ---------------------------------------------------------------------------
extern "C" void kernel_launch(void* const* d_in, const int* in_sizes, int n_in,
                              void* d_out, int out_size, void* d_ws, size_t ws_size,
                              hipStream_t stream) {
  const float* x   = (const float*)d_in[0];   // [N,128]
  const int*   ei  = (const int*)  d_in[1];   // [2,E] (JAX default int32)
  const float* ew  = (const float*)d_in[2];   // [E]
  const float* ef  = (const float*)d_in[3];   // [1024,64]
  const float* adj = (const float*)d_in[4];   // [1024,1024]
  const float* T   = (const float*)d_in[5];   // [N,1024]
  const float* W1  = (const float*)d_in[6];   // [128,64]
  const float* b1  = (const float*)d_in[7];
  const float* W2  = (const float*)d_in[8];   // [64,32]
  const float* b2  = (const float*)d_in[9];
  const float* Wn  = (const float*)d_in[10];  // [32,32]
  const float* We  = (const float*)d_in[11];  // [64,32]
  const float* bg  = (const float*)d_in[12];
  const float* Wc1 = (const float*)d_in[13];  // [32,32]
  const float* bc1 = (const float*)d_in[14];
  const float* Wc2 = (const float*)d_in[15];  // [32,2]
  const float* bc2 = (const float*)d_in[16];

  const int N  = in_sizes[0] / 128;   // 50000
  const int E  = in_sizes[1] / 2;     // 1,000,000
  const int NH = 1024;
  const int*   src = ei;
  const int*   dst = ei + E;

  // workspace layout (floats)
  float* ws   = (float*)d_ws;
  float* hx   = ws;                       // [N,64] -- reused for h2pre, shared
  float* out1 = hx   + (long)N * 64;      // [N,64]
  float* out2 = out1 + (long)N * 64;      // [N,32] -> relu in place -> h2
  float* nf   = out2 + (long)N * 32;      // [N,32]
  float* efWe = nf   + (long)N * 32;      // [1024,32]
  float* eh   = efWe + (long)NH * 32;     // [1024,32]

  // 0) zero all scatter accumulators (out1|out2|nf contiguous: N*128 floats)
  zero_kernel<<<2048, 256, 0, stream>>>(out1, (long)N * 128);

  // 1) hx = x @ W1 + b1                         [N,64]  (x streamed once, NT=4)
  {
    dim3 g((N + 15) / 16, 1);
    wmma_gemm_f32_kernel<4, false, true, false><<<g, 32, 0, stream>>>(
        x, 128, W1, 64, b1, hx, 64, N, 128);
  }
  // 2) out1[dst] += w * hx[src]                 (F=64 -> F4=16)
  {
    long total = (long)E * 16;
    edge_scatter_kernel<<<(unsigned)((total + 255) / 256), 256, 0, stream>>>(
        hx, src, dst, ew, out1, 16, E);
  }
  // 3) h2pre = relu(out1) @ W2 + b2             [N,32]  (stored in hx, NT=2)
  {
    dim3 g((N + 15) / 16, 1);
    wmma_gemm_f32_kernel<2, true, true, false><<<g, 32, 0, stream>>>(
        out1, 64, W2, 32, b2, hx, 32, N, 64);
  }
  // 4) out2[dst] += w * h2pre[src]              (F=32 -> F4=8)
  {
    long total = (long)E * 8;
    edge_scatter_kernel<<<(unsigned)((total + 255) / 256), 256, 0, stream>>>(
        hx, src, dst, ew, out2, 8, E);
  }
  // 5) h2 = relu(out2) in place
  relu_inplace_kernel<<<2048, 256, 0, stream>>>(out2, (long)N * 32);
  // 6) nf[dst] += h2[src]                       (unweighted)
  {
    long total = (long)E * 8;
    edge_scatter_kernel<<<(unsigned)((total + 255) / 256), 256, 0, stream>>>(
        out2, src, dst, nullptr, nf, 8, E);
  }
  // 7) efWe = edge_features @ We                [1024,32] (NT=2)
  {
    dim3 g(NH / 16, 1);
    wmma_gemm_f32_kernel<2, false, false, false><<<g, 32, 0, stream>>>(
        ef, 64, We, 32, nullptr, efWe, 32, NH, 64);
  }
  // 8) eh = adj_e @ efWe                        [1024,32] (NT=2)
  {
    dim3 g(NH / 16, 1);
    wmma_gemm_f32_kernel<2, false, false, false><<<g, 32, 0, stream>>>(
        adj, 1024, efWe, 32, nullptr, eh, 32, NH, 1024);
  }
  // 9) shared = relu(T @ eh + (nf/E) @ Wn + bg)  [N,32]  (stored in hx)
  {
    dim3 g((N + 15) / 16, 1);
    shared_fused_wmma_kernel<<<g, 32, 0, stream>>>(
        T, eh, nf, Wn, bg, 1.0f / (float)E, hx, N);
  }
  // 10) classifier + softmax -> d_out           [N,2]
  classifier_kernel<<<(N + 255) / 256, 256, 0, stream>>>(
      hx, Wc1, bc1, Wc2, bc2, (float*)d_out, N);
}